// TargetGuidance_43971875176871
// MI455X (gfx1250) — compile-verified
//
#include <hip/hip_runtime.h>
#include <math.h>

typedef __attribute__((ext_vector_type(2))) float v2f;
typedef __attribute__((ext_vector_type(8))) float v8f;

#define M_TGT   1024          // number of target points
#define SEQ_LEN 512           // points per batch row
#define TPB     256           // threads per block (8 wave32)
#define WPB     (TPB / 32)
#define NTILE   (M_TGT / 16)  // 64 B-tiles of 16 targets

// ---------------------------------------------------------------------------
// Kernel 1: replicate _augment_target on device, emit augmented target t'
// (2*M floats) and |t'|^2 (M floats) into workspace; zero the output scalar.
// One block of 1024 threads; LDS tree reduction for the mean segment length.
// ---------------------------------------------------------------------------
__global__ __launch_bounds__(M_TGT)
void tg_augment_kernel(const float* __restrict__ target,
                       float* __restrict__ aug,
                       float* __restrict__ tsq,
                       float* __restrict__ out) {
  __shared__ float red[M_TGT];
  __shared__ int   n_sh;
  const int tid = threadIdx.x;

  const float tx = target[2 * tid];
  const float ty = target[2 * tid + 1];

  float nrm = 0.0f;                       // ||t[i] - t[i-1]|| for i >= 1
  if (tid >= 1) {
    const float dx = tx - target[2 * (tid - 1)];
    const float dy = ty - target[2 * (tid - 1) + 1];
    nrm = sqrtf(dx * dx + dy * dy);
  }
  red[tid] = nrm;
  __syncthreads();
  for (int off = M_TGT / 2; off > 0; off >>= 1) {
    if (tid < off) red[tid] += red[tid + off];
    __syncthreads();
  }
  if (tid == 0) {
    const float dspace = red[0] / (float)(M_TGT - 1);
    const float t0x = target[0], t0y = target[1];
    const float d0 = sqrtf(t0x * t0x + t0y * t0y);
    float nf = floorf(d0 / (dspace + 1e-5f));
    nf = fminf(nf, 2.0e9f);               // guard int cast
    n_sh = (int)nf;
    out[0] = 0.0f;                        // accumulator for atomic adds
  }
  __syncthreads();

  const int n = n_sh;
  float ox, oy;
  if (n > 0) {
    if (tid < n) {
      const float inv = 1.0f / (float)n;  // n_safe == n here
      ox = (float)(tid + 1) * target[0] * inv;
      oy = (float)(tid + 1) * target[1] * inv;
    } else {
      int src = tid - n;                  // >= 0 in this branch
      if (src > M_TGT - 1) src = M_TGT - 1;
      ox = target[2 * src];
      oy = target[2 * src + 1];
    }
  } else {                                // n == 0 -> original target
    ox = tx; oy = ty;
  }
  aug[2 * tid]     = ox;
  aug[2 * tid + 1] = oy;
  tsq[tid]         = ox * ox + oy * oy;
}

// ---------------------------------------------------------------------------
// Kernel 2: WMMA distance kernel.
//   A (16x4, per wave):  row M = [-2px, -2py, 1, |p|^2]
//   B (4x16, per tile):  col N = [ tx,   ty, |t|^2, 1 ]
//   D = A*B  => exact squared distances, 16 points x 16 targets per WMMA.
// A layout (32-bit A 16x4): lanes 0-15 hold K=0,1 in v0,v1; lanes 16-31 K=2,3.
// B layout mirrors it: lanes 0-15 hold K=0,1; lanes 16-31 hold K=2,3.
// B operand lanes are pre-swizzled & interleaved into LDS so the inner loop
// is a single ds_load_b64 per tile feeding the WMMA operand pair directly.
// ---------------------------------------------------------------------------
__global__ __launch_bounds__(TPB)
void tg_dmin_wmma_kernel(const float* __restrict__ x,
                         const float* __restrict__ aug,
                         const float* __restrict__ tsq,
                         float* __restrict__ out) {
  __shared__ float2 sB[32 * NTILE];   // 16 KB: [tile*32 + lane] = {b0, b1}

  const int tid = threadIdx.x;

  // Stage pre-swizzled, interleaved B operands: entry e = tile*32 + lane.
  for (int e = tid; e < 32 * NTILE; e += TPB) {
    const int lane = e & 31;
    const int tile = e >> 5;
    const int tgt  = tile * 16 + (lane & 15);
    float2 b;
    if (lane < 16) { b.x = aug[2 * tgt]; b.y = aug[2 * tgt + 1]; } // K=0,1
    else           { b.x = tsq[tgt];     b.y = 1.0f;             } // K=2,3
    sB[e] = b;
  }
  __syncthreads();

  const int lane      = tid & 31;
  const int gwave     = blockIdx.x * WPB + (tid >> 5);
  const int pointBase = gwave * 16;
  const int p         = pointBase + (lane & 15);

  // x is (B,S,4); point p's xy at 16B-aligned offset 4*p.
  const float2 pxy = *(const float2*)(x + 4 * p);

  v2f a;
  if (lane < 16) { a.x = -2.0f * pxy.x;  a.y = -2.0f * pxy.y; }          // K=0,1
  else           { a.x = 1.0f;           a.y = pxy.x * pxy.x + pxy.y * pxy.y; } // K=2,3

  v8f rm;
#pragma unroll
  for (int k = 0; k < 8; ++k) rm[k] = 3.0e38f;
  const v8f cz = {};   // zero C operand

#pragma unroll 4
  for (int t = 0; t < NTILE; ++t) {
    const float2 bv = sB[t * 32 + lane];   // one ds_load_b64 per tile
    v2f b;
    b.x = bv.x;
    b.y = bv.y;
    // D = A x B + 0 : 16x16 tile of squared distances
    v8f d = __builtin_amdgcn_wmma_f32_16x16x4_f32(
        /*neg_a=*/false, a, /*neg_b=*/false, b,
        /*c_mod=*/(short)0, cz, /*reuse_a=*/false, /*reuse_b=*/false);
#pragma unroll
    for (int k = 0; k < 8; ++k) rm[k] = fminf(rm[k], d[k]);
  }

  // Per-row min over columns: columns live across the 16 lanes of each half,
  // and xor masks {1,2,4,8} never leave a 16-lane group.
#pragma unroll
  for (int msk = 1; msk <= 8; msk <<= 1) {
#pragma unroll
    for (int k = 0; k < 8; ++k)
      rm[k] = fminf(rm[k], __shfl_xor(rm[k], msk, 32));
  }

  // Lanes 0-15 now hold row-mins for M=0..7 (in rm[0..7]),
  // lanes 16-31 hold rows M=8..15.
  float s8 = 0.0f;
#pragma unroll
  for (int k = 0; k < 8; ++k) s8 += rm[k];
  const float tot = s8 + __shfl_xor(s8, 16, 32);   // sum of all 16 row-mins

  if (lane == 0) {
    // Exclude s==0 (row 0 of waves whose base starts a batch row).
    const bool first = (pointBase % SEQ_LEN) == 0;
    const float val  = tot - (first ? rm[0] : 0.0f);
    atomicAdd(out, val * (1.0f / (float)(SEQ_LEN - 1)));
  }
}

// ---------------------------------------------------------------------------
extern "C" void kernel_launch(void* const* d_in, const int* in_sizes, int n_in,
                              void* d_out, int out_size, void* d_ws, size_t ws_size,
                              hipStream_t stream) {
  const float* x      = (const float*)d_in[0];   // (64,512,4) f32
  const float* target = (const float*)d_in[1];   // (1024,2)   f32
  float* out = (float*)d_out;                    // scalar f32

  float* aug = (float*)d_ws;                     // 2*M_TGT floats
  float* tsq = aug + 2 * M_TGT;                  // M_TGT floats

  const int numPoints = in_sizes[0] / 4;         // 32768
  const int nWaves    = numPoints / 16;          // 2048
  const int nBlocks   = nWaves / WPB;            // 256

  hipLaunchKernelGGL(tg_augment_kernel, dim3(1), dim3(M_TGT), 0, stream,
                     target, aug, tsq, out);
  hipLaunchKernelGGL(tg_dmin_wmma_kernel, dim3(nBlocks), dim3(TPB), 0, stream,
                     x, aug, tsq, out);
}